// DynamicGEMModel_46858093199626
// MI455X (gfx1250) — compile-verified
//
#include <hip/hip_runtime.h>
#include <hip/hip_bf16.h>

typedef __attribute__((ext_vector_type(2))) float v2f;
typedef __attribute__((ext_vector_type(8))) float v8f;

#define FIN  128
#define HF   64
#define TOUT 16

__device__ __forceinline__ v8f wmma4(v2f a, v2f b, v8f c) {
  // D = A(16x4 f32) * B(4x16 f32) + C(16x16 f32)
  return __builtin_amdgcn_wmma_f32_16x16x4_f32(
      /*neg_a=*/false, a, /*neg_b=*/false, b,
      /*c_mod=*/(short)0, c, /*reuse_a=*/false, /*reuse_b=*/false);
}

__device__ __forceinline__ float sigmoidf(float x) {
  return 1.0f / (1.0f + expf(-x));
}

// ---------------- elementwise helpers ----------------

__global__ void k_fill(float* __restrict__ p, float v, int n) {
  int i = blockIdx.x * blockDim.x + threadIdx.x;
  if (i < n) p[i] = v;
}

__global__ void k_deg(const int* __restrict__ dst, float* __restrict__ deg, int e) {
  int i = blockIdx.x * blockDim.x + threadIdx.x;
  if (i < e) atomicAdd(&deg[dst[i]], 1.0f);
}

__global__ void k_rsqrt(float* __restrict__ d, int n) {
  int i = blockIdx.x * blockDim.x + threadIdx.x;
  if (i < n) d[i] = rsqrtf(d[i]);  // deg >= 1 always (self-loops)
}

// msg = dinv[src]*dinv[dst] * hw[src,f]  scattered into agg[dst,f]
__global__ void k_scatter(const int* __restrict__ src, const int* __restrict__ dst,
                          const float* __restrict__ dinv, const float* __restrict__ hw,
                          float* __restrict__ agg, int total /* E*64 */) {
  int i = blockIdx.x * blockDim.x + threadIdx.x;
  if (i >= total) return;
  int e = i >> 6;
  int f = i & (HF - 1);
  int s = src[e], d = dst[e];
  float nm = dinv[s] * dinv[d];
  atomicAdd(&agg[d * HF + f], nm * hw[s * HF + f]);
}

// h = relu(agg + dinv^2*hw (self-loop) + bias)   (in-place safe: out == agg)
__global__ void k_finalize(const float* __restrict__ agg, const float* __restrict__ hw,
                           const float* __restrict__ dinv, const float* __restrict__ bias,
                           float* __restrict__ out, int n) {
  int i = blockIdx.x * blockDim.x + threadIdx.x;
  if (i >= n * HF) return;
  int node = i >> 6;
  int f = i & (HF - 1);
  float dv = dinv[node];
  float v = agg[i] + dv * dv * hw[i] + bias[f];
  out[i] = fmaxf(v, 0.0f);
}

// ---------------- WMMA GEMM: C[nrows,NC] = A[nrows,K] @ B[K,NC] (+bias) ----------------
// Block = 256 thr = 8 waves; each wave computes a 32-row (2 M-tiles) x NC strip.
// B staged transposed+padded in LDS: Bsh[col*(K+2) + k] -> single ds_load_b64 frags,
// banks rotate by 2 per column (no conflicts).

template<int K, int NT>
__global__ void k_gemm(const float* __restrict__ A, const float* __restrict__ B,
                       const float* __restrict__ bias, float* __restrict__ C, int nrows) {
  const int NC  = NT * 16;
  const int KP  = K + 2;
  __shared__ float Bsh[NC * KP];

  // stage B transposed: Bsh[c*KP + k] = B[k*NC + c]
  for (int i = threadIdx.x; i < K * NC; i += blockDim.x) {
    int k = i / NC, c = i % NC;
    Bsh[c * KP + k] = B[i];
  }
  __syncthreads();

  int wave = (blockIdx.x * blockDim.x + threadIdx.x) >> 5;
  int lane = threadIdx.x & 31;
  int row0 = wave << 5;                 // 32 rows per wave
  int half = lane >> 4;
  int l16  = lane & 15;
  int ar0 = row0 + l16;       if (ar0 >= nrows) ar0 = nrows - 1;
  int ar1 = row0 + 16 + l16;  if (ar1 >= nrows) ar1 = nrows - 1;

  v8f zero = {};
  v8f acc[2][NT];
#pragma unroll
  for (int m = 0; m < 2; ++m)
#pragma unroll
    for (int t = 0; t < NT; ++t) acc[m][t] = zero;

  const float* ap0 = A + (size_t)ar0 * K + 2 * half;   // K index = k + 2*half + v
  const float* ap1 = A + (size_t)ar1 * K + 2 * half;
#pragma unroll 4
  for (int k = 0; k < K; k += 4) {
    v2f a0; a0.x = ap0[k]; a0.y = ap0[k + 1];
    v2f a1; a1.x = ap1[k]; a1.y = ap1[k + 1];
#pragma unroll
    for (int t = 0; t < NT; ++t) {
      int col = t * 16 + l16;
      v2f b;
      b.x = Bsh[col * KP + k + 2 * half];
      b.y = Bsh[col * KP + k + 2 * half + 1];
      acc[0][t] = wmma4(a0, b, acc[0][t]);
      acc[1][t] = wmma4(a1, b, acc[1][t]);
    }
  }

#pragma unroll
  for (int m = 0; m < 2; ++m)
#pragma unroll
    for (int t = 0; t < NT; ++t) {
      int col = t * 16 + l16;
      float badd = bias ? bias[col] : 0.0f;
#pragma unroll
      for (int g = 0; g < 8; ++g) {
        int r = row0 + m * 16 + g + 8 * half;   // C/D layout: vgpr g -> row g + 8*half
        if (r < nrows) C[(size_t)r * NC + col] = acc[m][t][g] + badd;
      }
    }
}

// ---------------- fused GRU (single step) ----------------
// gi_gate = h @ w_ih[gate].T ; gh_gate = h0 @ w_hh[gate].T
// Gate weight slices (64x64 each) staged in padded LDS per gate.

#define HFP (HF + 2)

__device__ __forceinline__ void gru_gate_lds(const float* __restrict__ h,
                                             const float* __restrict__ h0,
                                             const float* __restrict__ Wi,  // LDS [64*HFP]
                                             const float* __restrict__ Wh,  // LDS [64*HFP]
                                             int arow, int half, int l16,
                                             v8f gi[4], v8f gh[4]) {
  v8f zero = {};
#pragma unroll
  for (int t = 0; t < 4; ++t) { gi[t] = zero; gh[t] = zero; }
  const float* hp  = h  + (size_t)arow * HF + 2 * half;
  const float* h0p = h0 + (size_t)arow * HF + 2 * half;
#pragma unroll 4
  for (int k = 0; k < HF; k += 4) {
    v2f a;  a.x  = hp[k];  a.y  = hp[k + 1];
    v2f a0; a0.x = h0p[k]; a0.y = h0p[k + 1];
#pragma unroll
    for (int t = 0; t < 4; ++t) {
      int c = t * 16 + l16;            // output column within gate
      v2f bi, bh;
      bi.x = Wi[c * HFP + k + 2 * half];
      bi.y = Wi[c * HFP + k + 2 * half + 1];
      bh.x = Wh[c * HFP + k + 2 * half];
      bh.y = Wh[c * HFP + k + 2 * half + 1];
      gi[t] = wmma4(a,  bi, gi[t]);
      gh[t] = wmma4(a0, bh, gh[t]);
    }
  }
}

__global__ void k_gru(const float* __restrict__ h, const float* __restrict__ h0,
                      const float* __restrict__ w_ih, const float* __restrict__ w_hh,
                      const float* __restrict__ b_ih, const float* __restrict__ b_hh,
                      float* __restrict__ hnew, int nrows) {
  __shared__ float Wi[HF * HFP];
  __shared__ float Wh[HF * HFP];

  int wave = (blockIdx.x * blockDim.x + threadIdx.x) >> 5;
  int lane = threadIdx.x & 31;
  int row0 = wave << 4;
  int half = lane >> 4;
  int l16  = lane & 15;
  int arow = row0 + l16; if (arow >= nrows) arow = nrows - 1;

  v8f gi[4], gh[4], r_[4], z_[4];

#pragma unroll 1
  for (int gate = 0; gate < 3; ++gate) {
    __syncthreads();                   // previous gate done with LDS
    for (int i = threadIdx.x; i < HF * HF; i += blockDim.x) {
      int c = i / HF, k = i % HF;      // w row (within gate), k
      Wi[c * HFP + k] = w_ih[gate * HF * HF + i];
      Wh[c * HFP + k] = w_hh[gate * HF * HF + i];
    }
    __syncthreads();

    gru_gate_lds(h, h0, Wi, Wh, arow, half, l16, gi, gh);

    if (gate == 0) {
#pragma unroll
      for (int t = 0; t < 4; ++t) {
        int c = t * 16 + l16;
        float bi = b_ih[c], bh = b_hh[c];
#pragma unroll
        for (int g = 0; g < 8; ++g) r_[t][g] = sigmoidf(gi[t][g] + bi + gh[t][g] + bh);
      }
    } else if (gate == 1) {
#pragma unroll
      for (int t = 0; t < 4; ++t) {
        int c = t * 16 + l16;
        float bi = b_ih[HF + c], bh = b_hh[HF + c];
#pragma unroll
        for (int g = 0; g < 8; ++g) z_[t][g] = sigmoidf(gi[t][g] + bi + gh[t][g] + bh);
      }
    } else {
#pragma unroll
      for (int t = 0; t < 4; ++t) {
        int c = t * 16 + l16;
        float bi = b_ih[2 * HF + c], bh = b_hh[2 * HF + c];
#pragma unroll
        for (int g = 0; g < 8; ++g) {
          int r = row0 + g + 8 * half;
          int rr = r < nrows ? r : nrows - 1;
          float nv = tanhf(gi[t][g] + bi + r_[t][g] * (gh[t][g] + bh));
          float h0v = h0[(size_t)rr * HF + c];
          float zz = z_[t][g];
          float hv = (1.0f - zz) * nv + zz * h0v;
          if (r < nrows) hnew[(size_t)r * HF + c] = hv;
        }
      }
    }
  }
}

// ---------------- launcher ----------------

extern "C" void kernel_launch(void* const* d_in, const int* in_sizes, int n_in,
                              void* d_out, int out_size, void* d_ws, size_t ws_size,
                              hipStream_t stream) {
  const float* x   = (const float*)d_in[0];
  const int*  eidx = (const int*) d_in[1];
  const float* h0  = (const float*)d_in[2];
  const float* W1  = (const float*)d_in[3];
  const float* b1  = (const float*)d_in[4];
  const float* W2  = (const float*)d_in[5];
  const float* b2  = (const float*)d_in[6];
  const float* wih = (const float*)d_in[7];
  const float* whh = (const float*)d_in[8];
  const float* bih = (const float*)d_in[9];
  const float* bhh = (const float*)d_in[10];
  const float* Wp  = (const float*)d_in[11];
  const float* bp  = (const float*)d_in[12];

  const int N = in_sizes[0] / FIN;
  const int E = in_sizes[1] / 2;
  const int* src = eidx;
  const int* dst = eidx + E;

  float* wsf  = (float*)d_ws;
  float* dinv = wsf;                       // N
  float* hw   = dinv + N;                  // N*HF
  float* bufA = hw   + (size_t)N * HF;     // N*HF
  float* bufB = bufA + (size_t)N * HF;     // N*HF
  float* outp = (float*)d_out;             // N*TOUT
  float* hnew = outp + (size_t)N * TOUT;   // N*HF  (second tuple output)

  const int tpb = 256;
  const int NH  = N * HF;
  const int tot = E * HF;                                 // 204.8M < 2^31
  const int gemmBlocks = (((N + 31) / 32) + 7) / 8;       // 8 waves x 32 rows / block
  const int gruBlocks  = (((N + 15) / 16) + 7) / 8;       // 8 waves x 16 rows / block

  // degree -> dinv
  k_fill <<<(N + tpb - 1) / tpb, tpb, 0, stream>>>(dinv, 1.0f, N);   // self-loop
  k_deg  <<<(E + tpb - 1) / tpb, tpb, 0, stream>>>(dst, dinv, E);
  k_rsqrt<<<(N + tpb - 1) / tpb, tpb, 0, stream>>>(dinv, N);

  // layer 1: hw = x @ W1 ; agg = scatter(norm * hw[src]) ; h1 = relu(agg + selfloop + b1)
  k_gemm<FIN, HF / 16><<<gemmBlocks, 256, 0, stream>>>(x, W1, nullptr, hw, N);
  k_fill    <<<(NH + tpb - 1) / tpb, tpb, 0, stream>>>(bufA, 0.0f, NH);
  k_scatter <<<(tot + tpb - 1) / tpb, tpb, 0, stream>>>(src, dst, dinv, hw, bufA, tot);
  k_finalize<<<(NH + tpb - 1) / tpb, tpb, 0, stream>>>(bufA, hw, dinv, b1, bufA, N);

  // layer 2
  k_gemm<HF, HF / 16><<<gemmBlocks, 256, 0, stream>>>(bufA, W2, nullptr, hw, N);
  k_fill    <<<(NH + tpb - 1) / tpb, tpb, 0, stream>>>(bufB, 0.0f, NH);
  k_scatter <<<(tot + tpb - 1) / tpb, tpb, 0, stream>>>(src, dst, dinv, hw, bufB, tot);
  k_finalize<<<(NH + tpb - 1) / tpb, tpb, 0, stream>>>(bufB, hw, dinv, b2, bufB, N);

  // fused GRU step -> hnew (tuple output #2)
  k_gru<<<gruBlocks, 256, 0, stream>>>(bufB, h0, wih, whh, bih, bhh, hnew, N);

  // readout: out = hnew @ Wp + bp (tuple output #1)
  k_gemm<HF, TOUT / 16><<<gemmBlocks, 256, 0, stream>>>(hnew, Wp, bp, outp, N);
}